// SparseVoxelDetLoss_7284264534481
// MI455X (gfx1250) — compile-verified
//
#include <hip/hip_runtime.h>
#include <stdint.h>

#define MBLK 256
#define B_ 8
#define G_ 40
#define BG (B_ * G_)          // 320 (b,g) bins
#define INF_U 0x7F800000u     // +inf bits (dist^2 >= 0 -> uint order == float order)
#define NWAVE (MBLK / 32)     // 8 wave32 waves per block

// ---------------------------------------------------------------------------
// CDNA5 async global->LDS copy (ASYNCcnt path, CDNA5 ISA).
// Flat shared pointers on gfx1250 are {SHARED_BASE, lds_offset[31:0]}; hardware
// truncates to the low 32 bits for the LDS address, so a 32-bit cast of the
// generic pointer is the LDS byte address.
// ---------------------------------------------------------------------------
__device__ __forceinline__ uint32_t lds_off32(const void* p) {
    return (uint32_t)(uintptr_t)p;
}

__device__ __forceinline__ void async_copy_b128(uint32_t lds_byte_addr, const void* gsrc) {
    uint64_t ga = (uint64_t)(uintptr_t)gsrc;
    asm volatile("global_load_async_to_lds_b128 %0, %1, off"
                 :: "v"(lds_byte_addr), "v"(ga) : "memory");
}

__device__ __forceinline__ void wait_async_all() {
    asm volatile("s_wait_asynccnt 0" ::: "memory");
}

// dist^2 with no FMA contraction: computed identically in both passes so the
// pass-2 bit-equality test against the pass-1 segment-min is exact.
__device__ __forceinline__ float dist2_exact(float dx, float dy) {
    return __fadd_rn(__fmul_rn(dx, dx), __fmul_rn(dy, dy));
}

// ---------------------------------------------------------------------------
// Kernel 0: initialize the 320 min-dist bins and 320 any-match bins
// ---------------------------------------------------------------------------
__global__ void k_init(uint32_t* __restrict__ g_mind, int* __restrict__ g_any) {
    int i = threadIdx.x;
    if (i < BG) { g_mind[i] = INF_U; g_any[i] = 0; }
}

// ---------------------------------------------------------------------------
// Kernel 1: per-(b,g) segment reductions: any_match (OR), min finite dist^2.
// Both are order-invariant -> deterministic despite atomics.
// ---------------------------------------------------------------------------
__global__ __launch_bounds__(MBLK) void k_pass1(
    const int* __restrict__ idx, const float* __restrict__ gt,
    uint32_t* __restrict__ g_mind, int* __restrict__ g_any, int M)
{
    __shared__ alignas(16) float    s_box[BG * 4];
    __shared__ alignas(16) uint32_t s_mind[BG];
    __shared__ alignas(16) int      s_any[BG];

    const int tid = threadIdx.x;

    // Stage gt_boxes (1280 dwords = 320 x b128) via async global->LDS DMA.
    for (int c = tid; c < BG; c += MBLK)
        async_copy_b128(lds_off32(&s_box[c * 4]), gt + c * 4);
    for (int j = tid; j < BG; j += MBLK) { s_mind[j] = INF_U; s_any[j] = 0; }
    wait_async_all();
    __syncthreads();

    const int i = blockIdx.x * MBLK + tid;
    if (i < M) {
        const int   b  = idx[3 * i + 0];
        const float cy = (float)idx[3 * i + 1] * 4.0f + 2.0f;
        const float cx = (float)idx[3 * i + 2] * 4.0f + 2.0f;
        const float* bb = &s_box[b * G_ * 4];
        const int base = b * G_;
        #pragma unroll 4
        for (int g = 0; g < G_; ++g) {
            float x1 = bb[4*g+0], y1 = bb[4*g+1], x2 = bb[4*g+2], y2 = bb[4*g+3];
            float l = cx - x1, t = cy - y1, r = x2 - cx, bo = y2 - cy;
            float mn  = fminf(fminf(l, t), fminf(r, bo));
            float gcx = (x1 + x2) * 0.5f, gcy = (y1 + y2) * 0.5f;
            float dx = cx - gcx, dy = cy - gcy;
            bool match = (mn > 0.0f) && (fabsf(dx) < 6.0f) && (fabsf(dy) < 6.0f);
            float d2 = dist2_exact(dx, dy);
            if (match)        atomicOr(&s_any[base + g], 1);              // ds_or_b32
            if (d2 < 1024.0f) atomicMin(&s_mind[base + g],                // ds_min_u32
                                        __float_as_uint(d2));
        }
    }
    __syncthreads();

    // One global atomic per touched bin per block.
    for (int j = tid; j < BG; j += MBLK) {
        if (s_any[j]) atomicOr(&g_any[j], 1);
        uint32_t m = s_mind[j];
        if (m != INF_U) atomicMin(&g_mind[j], m);
    }
}

// ---------------------------------------------------------------------------
// Kernel 2: assignment + losses; deterministic per-block partial sums.
// partials[blockIdx*4 + {0,1,2,3}] = {cls_sum, reg_sum, ctr_sum, num_pos}
// ---------------------------------------------------------------------------
__global__ __launch_bounds__(MBLK) void k_pass2(
    const float* __restrict__ cls_logits, const float* __restrict__ ltrb_pred,
    const float* __restrict__ ctr_logits, const int* __restrict__ idx,
    const float* __restrict__ gt,
    const uint32_t* __restrict__ g_mind, const int* __restrict__ g_any,
    float* __restrict__ partials, int M)
{
    __shared__ alignas(16) float    s_box[BG * 4];
    __shared__ alignas(16) uint32_t s_mind[BG];
    __shared__ alignas(16) int      s_any[BG];
    __shared__ float s_wave[NWAVE][4];

    const int tid = threadIdx.x;

    for (int c = tid; c < BG; c += MBLK)
        async_copy_b128(lds_off32(&s_box[c * 4]), gt + c * 4);
    for (int c = tid; c < BG / 4; c += MBLK) {   // 80 b128 chunks each
        async_copy_b128(lds_off32(&s_mind[c * 4]), g_mind + c * 4);
        async_copy_b128(lds_off32(&s_any[c * 4]),  g_any  + c * 4);
    }
    wait_async_all();
    __syncthreads();

    const int i = blockIdx.x * MBLK + tid;
    float c_cls = 0.0f, c_reg = 0.0f, c_ctr = 0.0f, c_pos = 0.0f;

    if (i < M) {
        const int   b  = idx[3 * i + 0];
        const float cy = (float)idx[3 * i + 1] * 4.0f + 2.0f;
        const float cx = (float)idx[3 * i + 2] * 4.0f + 2.0f;
        const float* bb = &s_box[b * G_ * 4];
        const int base = b * G_;

        float bestArea = __uint_as_float(INF_U);
        float tl = 0.f, tt = 0.f, tr = 0.f, tb = 0.f;
        bool pos = false;

        #pragma unroll 4
        for (int g = 0; g < G_; ++g) {
            float x1 = bb[4*g+0], y1 = bb[4*g+1], x2 = bb[4*g+2], y2 = bb[4*g+3];
            float l = cx - x1, t = cy - y1, r = x2 - cx, bo = y2 - cy;
            float mn  = fminf(fminf(l, t), fminf(r, bo));
            float gcx = (x1 + x2) * 0.5f, gcy = (y1 + y2) * 0.5f;
            float dx = cx - gcx, dy = cy - gcy;
            bool match0 = (mn > 0.0f) && (fabsf(dx) < 6.0f) && (fabsf(dy) < 6.0f);
            float d2 = dist2_exact(dx, dy);
            uint32_t du  = (d2 < 1024.0f) ? __float_as_uint(d2) : INF_U;
            uint32_t mdu = s_mind[base + g];
            bool need_fb = (s_any[base + g] == 0) && (mdu < INF_U);
            bool nearest = (du == mdu) && (du < INF_U);
            if (match0 || (nearest && need_fb)) {
                float area = fmaxf(x2 - x1, 0.0f) * fmaxf(y2 - y1, 0.0f);
                if (area < bestArea) {                 // strict <: first-min tie-break
                    bestArea = area; tl = l; tt = t; tr = r; tb = bo;
                }
                pos = true;
            }
        }

        // Focal classification term (every point contributes).
        float x    = cls_logits[i];
        float clsT = pos ? 1.0f : 0.0f;
        float ce   = fmaxf(x, 0.0f) - x * clsT + log1pf(expf(-fabsf(x)));
        float p    = 1.0f / (1.0f + expf(-x));
        float p_t  = pos ? p : (1.0f - p);
        float a_t  = pos ? 0.25f : 0.75f;
        float om   = 1.0f - p_t;
        c_cls = a_t * om * om * ce;

        if (pos) {
            tl = fmaxf(tl, 0.01f); tt = fmaxf(tt, 0.01f);
            tr = fmaxf(tr, 0.01f); tb = fmaxf(tb, 0.01f);
            float lr  = fminf(tl, tr) / fmaxf(fmaxf(tl, tr), 1e-6f);
            float tbr = fminf(tt, tb) / fmaxf(fmaxf(tt, tb), 1e-6f);
            float ctrT = sqrtf(fmaxf(lr * tbr, 0.0f));

            const float4 lp = *(const float4*)(ltrb_pred + 4 * i);
            float pl = lp.x, pt = lp.y, pr = lp.z, pb = lp.w;
            float pa = fmaxf(pl + pr, 0.0f) * fmaxf(pt + pb, 0.0f);
            float ta = fmaxf(tl + tr, 0.0f) * fmaxf(tt + tb, 0.0f);
            float iw = fmaxf(fminf(pl, tl) + fminf(pr, tr), 0.0f);
            float ih = fmaxf(fminf(pt, tt) + fminf(pb, tb), 0.0f);
            float inter = iw * ih;
            float uni   = pa + ta - inter + 1e-7f;
            float iou   = inter / uni;
            float ew = fmaxf(fmaxf(pl, tl) + fmaxf(pr, tr), 0.0f);
            float eh = fmaxf(fmaxf(pt, tt) + fmaxf(pb, tb), 0.0f);
            float enc = ew * eh + 1e-7f;
            float giou = iou - (enc - uni) / enc;
            c_reg = (1.0f - giou) * ctrT;

            float cl = ctr_logits[i];
            c_ctr = fmaxf(cl, 0.0f) - cl * ctrT + log1pf(expf(-fabsf(cl)));
            c_pos = 1.0f;
        }
    }

    // Deterministic block reduction: wave32 shuffle tree, then fixed-order
    // fold of the 8 per-wave partials (no float atomics anywhere).
    #pragma unroll
    for (int off = 16; off > 0; off >>= 1) {
        c_cls += __shfl_down(c_cls, off, 32);
        c_reg += __shfl_down(c_reg, off, 32);
        c_ctr += __shfl_down(c_ctr, off, 32);
        c_pos += __shfl_down(c_pos, off, 32);
    }
    const int wave = tid >> 5;
    if ((tid & 31) == 0) {
        s_wave[wave][0] = c_cls;
        s_wave[wave][1] = c_reg;
        s_wave[wave][2] = c_ctr;
        s_wave[wave][3] = c_pos;
    }
    __syncthreads();
    if (tid < 4) {
        float s = 0.0f;
        #pragma unroll
        for (int w = 0; w < NWAVE; ++w) s += s_wave[w][tid];
        partials[(size_t)blockIdx.x * 4 + tid] = s;
    }
}

// ---------------------------------------------------------------------------
// Kernel 3: deterministic final reduction over block partials + 5 outputs
// ---------------------------------------------------------------------------
#define FBLK 256
__global__ __launch_bounds__(FBLK) void k_final(const float* __restrict__ partials,
                                                int nblocks, float* __restrict__ out) {
    __shared__ float s_w[FBLK / 32][4];
    float a0 = 0.f, a1 = 0.f, a2 = 0.f, a3 = 0.f;
    for (int bi = threadIdx.x; bi < nblocks; bi += FBLK) {
        a0 += partials[4 * bi + 0];
        a1 += partials[4 * bi + 1];
        a2 += partials[4 * bi + 2];
        a3 += partials[4 * bi + 3];
    }
    #pragma unroll
    for (int off = 16; off > 0; off >>= 1) {
        a0 += __shfl_down(a0, off, 32);
        a1 += __shfl_down(a1, off, 32);
        a2 += __shfl_down(a2, off, 32);
        a3 += __shfl_down(a3, off, 32);
    }
    const int wave = threadIdx.x >> 5;
    if ((threadIdx.x & 31) == 0) {
        s_w[wave][0] = a0; s_w[wave][1] = a1; s_w[wave][2] = a2; s_w[wave][3] = a3;
    }
    __syncthreads();
    if (threadIdx.x == 0) {
        float s0 = 0.f, s1 = 0.f, s2 = 0.f, s3 = 0.f;
        #pragma unroll
        for (int w = 0; w < FBLK / 32; ++w) {
            s0 += s_w[w][0]; s1 += s_w[w][1]; s2 += s_w[w][2]; s3 += s_w[w][3];
        }
        float np  = fmaxf(s3, 1.0f);
        float cls = s0 / np;
        float reg = s1 / np;
        float ctr = s2 / np;
        out[0] = cls + 2.0f * reg + ctr;   // CLS_W=1, REG_W=2, CTR_W=1
        out[1] = cls;
        out[2] = reg;
        out[3] = ctr;
        out[4] = np;
    }
}

extern "C" void kernel_launch(void* const* d_in, const int* in_sizes, int n_in,
                              void* d_out, int out_size, void* d_ws, size_t ws_size,
                              hipStream_t stream) {
    const float* cls  = (const float*)d_in[0];   // (M,)
    const float* ltrb = (const float*)d_in[1];   // (M,4)
    const float* ctr  = (const float*)d_in[2];   // (M,)
    const float* gt   = (const float*)d_in[3];   // (B,G,4)
    const int*   idx  = (const int*)d_in[4];     // (M,3)
    const int M = in_sizes[0];

    // Workspace layout (16B-aligned segments):
    //   [0,1280)      : uint32 min_d[320]
    //   [1280,2560)   : int    any_match[320]
    //   [2560, ...)   : float  partials[nblocks*4]
    uint32_t* g_mind   = (uint32_t*)d_ws;
    int*      g_any    = (int*)((char*)d_ws + 1280);
    float*    g_parts  = (float*)((char*)d_ws + 2560);

    const int blocks = (M + MBLK - 1) / MBLK;

    k_init <<<1, 512, 0, stream>>>(g_mind, g_any);
    k_pass1<<<blocks, MBLK, 0, stream>>>(idx, gt, g_mind, g_any, M);
    k_pass2<<<blocks, MBLK, 0, stream>>>(cls, ltrb, ctr, idx, gt,
                                         g_mind, g_any, g_parts, M);
    k_final<<<1, FBLK, 0, stream>>>(g_parts, blocks, (float*)d_out);
}